// SGP4Layer_5016521802243
// MI455X (gfx1250) — compile-verified
//
#include <hip/hip_runtime.h>
#include <math.h>

// ---------------------------------------------------------------------------
// SGP4 near-earth propagation, fp32, one satellite per lane.
// CDNA5 (gfx1250) specifics used:
//   * TDM tensor_load_to_lds to stage each wave's 32x7-float param slice into
//     LDS (descriptor per CDNA5 ISA D# layout), followed by
//     s_wait_tensorcnt 0. One TDM op per wave; stride-7 LDS reads are
//     bank-conflict free across 32 lanes (gcd(7,64)=1).
//   * global_prefetch (via __builtin_prefetch) to warm L2 for the next block.
//   * wave32 sizing: 256-thread blocks = 8 waves.
// VALU-bound workload: all trig uses a custom branch-free bounded-range
// sincos (3-step FMA Cody-Waite + Cephes minimax, ~1 ulp for |x| << 1e5 —
// SGP4 angles here are < ~1e3 rad); x^(2/3) uses hardware v_log/v_exp;
// fmod/powf eliminated.
// ---------------------------------------------------------------------------

typedef unsigned int u32;
typedef unsigned long long u64;
typedef __attribute__((ext_vector_type(4))) u32 v4u;
typedef __attribute__((ext_vector_type(8))) int v8i;
typedef __attribute__((ext_vector_type(4))) int v4i;

#if defined(__HIP_DEVICE_COMPILE__) && defined(__gfx1250__) && \
    __has_builtin(__builtin_amdgcn_tensor_load_to_lds) && \
    __has_builtin(__builtin_amdgcn_s_wait_tensorcnt)
#define CDNA5_HAVE_TDM 1
#else
#define CDNA5_HAVE_TDM 0
#endif

#define BLK 256

#define TWOPI_F   6.28318530717958647692f
#define INV2PI_F  0.15915494309189533577f
#define RE_F      6378.137f
#define J2_F      1.08262998905e-3f
#define J4_F      (-1.61098761e-6f)
#define J3OJ2_F   ((float)(-2.53215306e-6 / 1.08262998905e-3))
#define X2O3_F    (2.0f / 3.0f)
#define TEMP4_F   1.5e-12f

// Branch-free sincos for bounded arguments (|x| up to ~1e5 rad).
// 3-step FMA Cody-Waite reduction by pi/2, Cephes minimax polynomials on
// [-pi/4, pi/4], quadrant fix-up via selects. ~22 VALU ops, no SALU branches.
__device__ __forceinline__ void fast_sincosf(float x, float* sout, float* cout) {
    float q = rintf(x * 0.63661977236758134308f);          // round(x / (pi/2))
    float r = fmaf(q, -1.57079625129699707031e+00f, x);    // pi/2 hi
    r = fmaf(q, -7.54978941586159635335e-08f, r);          // pi/2 mid
    r = fmaf(q, -5.39030295341742356449e-15f, r);          // pi/2 lo
    int n = (int)q;
    float z = r * r;
    // sin poly (Cephes sinf)
    float s = fmaf(-1.9515295891e-4f, z, 8.3321608736e-3f);
    s = fmaf(s, z, -1.6666654611e-1f);
    s = fmaf(s * z, r, r);
    // cos poly (Cephes cosf)
    float w = z * z;
    float c = fmaf(2.443315711809948e-5f, z, -1.388731625493765e-3f);
    c = fmaf(c, z, 4.166664568298827e-2f);
    c = fmaf(c, w, fmaf(-0.5f, z, 1.0f));
    // quadrant: n mod 4 (two's complement & works for negative n)
    float sv = (n & 1) ? c : s;
    float cv = (n & 1) ? s : c;
    sv = (n & 2) ? -sv : sv;
    cv = ((n + 1) & 2) ? -cv : cv;
    *sout = sv;
    *cout = cv;
}
__device__ __forceinline__ float fast_sinf(float x) {
    float s, c; fast_sincosf(x, &s, &c); return s;   // cos poly DCE'd
}
__device__ __forceinline__ float fast_cosf(float x) {
    float s, c; fast_sincosf(x, &s, &c); return c;   // sin poly DCE'd
}

// x^(2/3) for well-conditioned x (~[1.2, 1.7] here): hardware log2/exp2
// (v_log_f32 / v_exp_f32, ~1 ulp each) -> ~1.5e-7 relative error, on par
// with XLA's exp/log-based pow lowering.
__device__ __forceinline__ float pow23f(float x) {
#if __has_builtin(__builtin_amdgcn_logf) && __has_builtin(__builtin_amdgcn_exp2f)
    // v_log_f32 is log2; v_exp_f32 is exp2
    return __builtin_amdgcn_exp2f(0.66666666666666667f * __builtin_amdgcn_logf(x));
#else
    return exp2f(0.66666666666666667f * log2f(x));
#endif
}

// Floor-mod into [0, 2*pi). Arguments here are bounded (|x| < ~1e3 rad).
// Matches jnp.mod floor semantics (non-negative result for positive divisor).
__device__ __forceinline__ float mod2pi(float x) {
    float k = floorf(x * INV2PI_F);
    float r = fmaf(-k, TWOPI_F, x);
    if (r < 0.0f) r += TWOPI_F;
    else if (r >= TWOPI_F) r -= TWOPI_F;
    return r;
}

__global__ __launch_bounds__(BLK) void sgp4_kernel(
    const float* __restrict__ params,   // (N,7)
    const float* __restrict__ tmin,     // (N,)
    float* __restrict__ out,            // pos (N,3) flat, then vel (N,3) flat
    int N)
{
    __shared__ float sp[BLK * 7];

    const int tid  = threadIdx.x;
    const int base = blockIdx.x * BLK;
    const int idx  = base + tid;

    // Warm L2 for the next block's params (192MB L2 >> 28MB input).
    if (base + BLK < N) {
        __builtin_prefetch(params + (size_t)(base + BLK) * 7, 0, 1);
    }

#if CDNA5_HAVE_TDM
    {
        // One TDM descriptor per wave: 1-D tile of 224 fp32 (this wave's
        // 32 satellites x 7 params) from global into this wave's LDS slice.
        const u32 waveId    = (u32)(tid >> 5);
        const u64 elemStart = (u64)base * 7u + (u64)waveId * 224u;
        const u64 gaddr     = (u64)(uintptr_t)params + elemStart * 4u;
        long long remain    = (long long)N * 7 - (long long)elemStart;
        u32 tdim0 = (remain <= 0) ? 1u
                  : (remain > 2147483647LL ? 2147483647u : (u32)remain);
        // Flat LDS addresses carry the byte offset in the low 32 bits.
        const u32 ldsaddr = ((u32)(uintptr_t)(&sp[0])) + waveId * 224u * 4u;

        v4u g0;
        g0[0] = 1u;                                            // count=1 (valid), user mode
        g0[1] = ldsaddr;                                       // lds_addr (bytes)
        g0[2] = (u32)(gaddr & 0xFFFFFFFFull);                  // global_addr[31:0]
        g0[3] = (u32)((gaddr >> 32) & 0x01FFFFFFull)           // global_addr[56:32]
              | (2u << 30);                                    // type=2 ("image")

        v8i g1;
        g1[0] = (int)(2u << 16);                               // wg_mask=0, data_size=4B
        g1[1] = (int)((tdim0 & 0xFFFFu) << 16);                // tensor_dim0[15:0]
        g1[2] = (int)(((tdim0 >> 16) & 0xFFFFu)                // tensor_dim0[31:16]
              | (1u << 16));                                   // tensor_dim1 = 1
        g1[3] = (int)(224u << 16);                             // tile_dim0 = 224
        g1[4] = 0;                                             // tile_dim1=0, tile_dim2=0
        g1[5] = (int)tdim0;                                    // tensor_dim0_stride lo
        g1[6] = 0;
        g1[7] = 0;

        v4i gz = {0, 0, 0, 0};
#if __clang_major__ >= 23
        v8i gz8 = {0, 0, 0, 0, 0, 0, 0, 0};
        __builtin_amdgcn_tensor_load_to_lds(g0, g1, gz, gz, gz8, 0);
#else
        __builtin_amdgcn_tensor_load_to_lds(g0, g1, gz, gz, 0);
#endif
        __builtin_amdgcn_s_wait_tensorcnt(0);
        // Each lane only reads its own wave's slice; no workgroup barrier needed.
    }
#else
    // Fallback: cooperative coalesced staging.
    for (int i = tid; i < BLK * 7; i += BLK) {
        long long g = (long long)base * 7 + i;
        sp[i] = (g < (long long)N * 7) ? params[g] : 0.0f;
    }
    __syncthreads();
#endif

    if (idx >= N) return;

    const float XKE_F = (float)(60.0 /
        __builtin_sqrt(6378.137 * 6378.137 * 6378.137 / 398600.5));
    const float qzms2t = (float)((42.0 / 6378.137) * (42.0 / 6378.137) *
                                 (42.0 / 6378.137) * (42.0 / 6378.137));
    const float ss = 78.0f / RE_F + 1.0f;

    const float* p = &sp[tid * 7];
    float n_kozai = p[0];
    float ecco    = p[1];
    float inclo   = p[2];
    float nodeo   = p[3];
    float argpo   = p[4];
    float mo      = p[5];
    float bstar   = p[6];
    float t = tmin[idx];

    // ----------------------------- sgp4init ------------------------------
    float eccsq  = ecco * ecco;
    float omeosq = 1.0f - eccsq;
    float rteosq = sqrtf(omeosq);
    float sinio, cosio;
    fast_sincosf(inclo, &sinio, &cosio);
    float cosio2 = cosio * cosio;
    float ak   = pow23f(XKE_F / n_kozai);
    float d1   = 0.75f * J2_F * (3.0f * cosio2 - 1.0f) / (rteosq * omeosq);
    float del_ = d1 / (ak * ak);
    float adel = ak * (1.0f - del_ * del_ -
                       del_ * (1.0f / 3.0f + 134.0f * del_ * del_ / 81.0f));
    del_ = d1 / (adel * adel);
    float no_unkozai = n_kozai / (1.0f + del_);
    float ao    = pow23f(XKE_F / no_unkozai);
    float po    = ao * omeosq;
    float con42 = 1.0f - 5.0f * cosio2;
    float con41 = -con42 - 2.0f * cosio2;
    float posq  = po * po;
    float rp    = ao * (1.0f - ecco);

    bool  isimp  = rp < (220.0f / RE_F + 1.0f);
    float perige = (rp - 1.0f) * RE_F;
    float sfour_low = (perige < 98.0f) ? 20.0f : perige - 78.0f;
    float qs = (120.0f - sfour_low) / RE_F;
    float qs2 = qs * qs;
    float qzms24_low = qs2 * qs2;
    bool  low    = perige < 156.0f;
    float sfour  = low ? sfour_low / RE_F + 1.0f : ss;
    float qzms24 = low ? qzms24_low : qzms2t;

    float pinvsq = 1.0f / posq;
    float tsi    = 1.0f / (ao - sfour);
    float eta    = ao * ecco * tsi;
    float etasq  = eta * eta;
    float eeta   = ecco * eta;
    float psisq  = fabsf(1.0f - etasq);
    float tsi2   = tsi * tsi;
    float coef   = qzms24 * (tsi2 * tsi2);
    // psisq^3.5 = psisq^3 * sqrt(psisq): exact-ish, no powf
    float psisq3 = psisq * psisq * psisq;
    float coef1  = coef / (psisq3 * sqrtf(psisq));
    float cc2 = coef1 * no_unkozai *
        (ao * (1.0f + 1.5f * etasq + eeta * (4.0f + etasq)) +
         0.375f * J2_F * tsi / psisq * con41 *
             (8.0f + 3.0f * etasq * (8.0f + etasq)));
    float cc1 = bstar * cc2;
    float safe_e = fmaxf(ecco, 1e-4f);
    float cc3 = (ecco > 1e-4f)
        ? -2.0f * coef * tsi * J3OJ2_F * no_unkozai * sinio / safe_e : 0.0f;
    float x1mth2 = 1.0f - cosio2;
    // cos(argpo) once; cos(2*argpo) via double-angle
    float cargpo  = fast_cosf(argpo);
    float cargpo2 = 2.0f * cargpo * cargpo - 1.0f;
    float cc4 = 2.0f * no_unkozai * coef1 * ao * omeosq *
        (eta * (2.0f + 0.5f * etasq) + ecco * (0.5f + 2.0f * etasq) -
         J2_F * tsi / (ao * psisq) *
             (-3.0f * con41 * (1.0f - 2.0f * eeta + etasq * (1.5f - 0.5f * eeta)) +
              0.75f * x1mth2 * (2.0f * etasq - eeta * (1.0f + etasq)) * cargpo2));
    float cc5 = 2.0f * coef1 * ao * omeosq *
        (1.0f + 2.75f * (etasq + eeta) + eeta * etasq);
    float cosio4 = cosio2 * cosio2;
    float temp1 = 1.5f * J2_F * pinvsq * no_unkozai;
    float temp2 = 0.5f * temp1 * J2_F * pinvsq;
    float temp3 = -0.46875f * J4_F * pinvsq * pinvsq * no_unkozai;
    float mdot = no_unkozai + 0.5f * temp1 * rteosq * con41 +
        0.0625f * temp2 * rteosq * (13.0f - 78.0f * cosio2 + 137.0f * cosio4);
    float argpdot = -0.5f * temp1 * con42 +
        0.0625f * temp2 * (7.0f - 114.0f * cosio2 + 395.0f * cosio4) +
        temp3 * (3.0f - 36.0f * cosio2 + 49.0f * cosio4);
    float xhdot1  = -temp1 * cosio;
    float nodedot = xhdot1 +
        (0.5f * temp2 * (4.0f - 19.0f * cosio2) +
         2.0f * temp3 * (3.0f - 7.0f * cosio2)) * cosio;
    float omgcof = bstar * cc3 * cargpo;
    float safe_eeta = (fabsf(eeta) > 1e-12f) ? eeta : 1.0f;
    float xmcof = (ecco > 1e-4f) ? -X2O3_F * coef * bstar / safe_eeta : 0.0f;
    float nodecf = 3.5f * omeosq * xhdot1 * cc1;
    float t2cof = 1.5f * cc1;
    float denom = (fabsf(1.0f + cosio) > 1.5e-12f) ? (1.0f + cosio) : TEMP4_F;
    float xlcof = -0.25f * J3OJ2_F * sinio * (3.0f + 5.0f * cosio) / denom;
    float aycof = -0.5f * J3OJ2_F * sinio;
    float smo, cmo;
    fast_sincosf(mo, &smo, &cmo);
    float dmo   = 1.0f + eta * cmo;
    float delmo = dmo * dmo * dmo;
    float sinmao = smo;
    float x7thm1 = 7.0f * cosio2 - 1.0f;
    // non-isimp drag coefficients (computed unconditionally, selected below)
    float cc1sq = cc1 * cc1;
    float d2c = 4.0f * ao * tsi * cc1sq;
    float tmpc = d2c * tsi * cc1 / 3.0f;
    float d3c = (17.0f * ao + sfour) * tmpc;
    float d4c = 0.5f * tmpc * ao * tsi * (221.0f * ao + 31.0f * sfour) * cc1;
    float t3cof = d2c + 2.0f * cc1sq;
    float t4cof = 0.25f * (3.0f * d3c + cc1 * (12.0f * d2c + 10.0f * cc1sq));
    float t5cof = 0.2f * (3.0f * d4c + 12.0f * cc1 * d3c + 6.0f * d2c * d2c +
                          15.0f * cc1sq * (2.0f * d2c + cc1sq));

    // ---------------------------- propagation ----------------------------
    float xmdf   = mo + mdot * t;
    float argpdf = argpo + argpdot * t;
    float nodedf = nodeo + nodedot * t;
    float t2 = t * t;
    float nodem  = nodedf + nodecf * t2;
    float tempa0 = 1.0f - cc1 * t;
    float tempe0 = bstar * cc4 * t;
    float templ0 = t2cof * t2;
    float delomg = omgcof * t;
    float dcx  = 1.0f + eta * fast_cosf(xmdf);
    float delm = xmcof * (dcx * dcx * dcx - delmo);
    float per  = delomg + delm;
    float mm    = isimp ? xmdf : xmdf + per;
    float argpm = isimp ? argpdf : argpdf - per;
    float t3 = t2 * t;
    float t4 = t3 * t;
    float tempa = isimp ? tempa0 : tempa0 - d2c * t2 - d3c * t3 - d4c * t4;
    float tempe = isimp ? tempe0
                        : tempe0 + bstar * cc5 * (fast_sinf(mm) - sinmao);
    float templ = isimp ? templ0
                        : templ0 + t3cof * t3 + t4cof * t4 + t5cof * t4 * t;

    float am = ao * tempa * tempa;           // ao == (XKE/no_unkozai)^(2/3)
    float nm = XKE_F / (am * sqrtf(am));     // XKE / am^1.5
    float em = fmaxf(ecco - tempe, 1e-6f);
    mm = mm + no_unkozai * templ;
    float xlm = mm + argpm + nodem;
    nodem = mod2pi(nodem);
    argpm = mod2pi(argpm);
    xlm   = mod2pi(xlm);
    mm    = mod2pi(xlm - argpm - nodem);

    float sinim = sinio;
    float cosim = cosio;
    float sargpm, cargpm;
    fast_sincosf(argpm, &sargpm, &cargpm);
    float axnl = em * cargpm;
    float tinv = 1.0f / (am * (1.0f - em * em));
    float aynl = em * sargpm + tinv * aycof;
    float xl = mm + argpm + nodem + tinv * xlcof * axnl;
    float u = mod2pi(xl - nodem);

    // Newton-Raphson Kepler solve, fixed 10 iterations, clamped step.
    float eo1 = u;
#pragma unroll 1
    for (int it = 0; it < 10; ++it) {
        float se, ce;
        fast_sincosf(eo1, &se, &ce);
        float tem5 = (u - aynl * ce + axnl * se - eo1) /
                     (1.0f - ce * axnl - se * aynl);
        tem5 = fminf(0.95f, fmaxf(-0.95f, tem5));
        eo1 += tem5;
    }

    float sineo1, coseo1;
    fast_sincosf(eo1, &sineo1, &coseo1);
    float ecose = axnl * coseo1 + aynl * sineo1;
    float esine = axnl * sineo1 - aynl * coseo1;
    float el2 = axnl * axnl + aynl * aynl;
    float pl = am * (1.0f - el2);
    float rl = am * (1.0f - ecose);
    float rdotl  = sqrtf(am) * esine / rl;
    float rvdotl = sqrtf(pl) / rl;
    float betal  = sqrtf(1.0f - el2);
    float tq = esine / (1.0f + betal);
    float sinu = am / rl * (sineo1 - aynl - axnl * tq);
    float cosu = am / rl * (coseo1 - axnl + aynl * tq);
    float su = atan2f(sinu, cosu);
    float sin2u = 2.0f * cosu * sinu;
    float cos2u = 1.0f - 2.0f * sinu * sinu;
    float pli = 1.0f / pl;
    float tb = 0.5f * J2_F * pli;
    float tc = tb * pli;
    // short-period periodics
    float mrt = rl * (1.0f - 1.5f * tc * betal * con41) +
                0.5f * tb * x1mth2 * cos2u;
    su = su - 0.25f * tc * x7thm1 * sin2u;
    float xnode = nodem + 1.5f * tc * cosim * sin2u;
    float xinc  = inclo + 1.5f * tc * cosim * sinim * cos2u;
    float mvt   = rdotl - nm * tb * x1mth2 * sin2u / XKE_F;
    float rvdot = rvdotl + nm * tb * (x1mth2 * cos2u + 1.5f * con41) / XKE_F;

    float sinsu, cossu;
    fast_sincosf(su, &sinsu, &cossu);
    float snod, cnod;
    fast_sincosf(xnode, &snod, &cnod);
    float sini, cosi;
    fast_sincosf(xinc, &sini, &cosi);
    float xmx = -snod * cosi;
    float xmy =  cnod * cosi;
    float ux = xmx * sinsu + cnod * cossu;
    float uy = xmy * sinsu + snod * cossu;
    float uz = sini * sinsu;
    float vx = xmx * cossu - cnod * sinsu;
    float vy = xmy * cossu - snod * sinsu;
    float vz = sini * cossu;

    const float vkps = RE_F * XKE_F / 60.0f;
    float r = mrt * RE_F;

    size_t pbase = (size_t)3 * (size_t)idx;
    size_t vbase = (size_t)3 * (size_t)N + pbase;
    out[pbase + 0] = r * ux;
    out[pbase + 1] = r * uy;
    out[pbase + 2] = r * uz;
    out[vbase + 0] = vkps * (mvt * ux + rvdot * vx);
    out[vbase + 1] = vkps * (mvt * uy + rvdot * vy);
    out[vbase + 2] = vkps * (mvt * uz + rvdot * vz);
}

extern "C" void kernel_launch(void* const* d_in, const int* in_sizes, int n_in,
                              void* d_out, int out_size, void* d_ws, size_t ws_size,
                              hipStream_t stream) {
    (void)n_in; (void)out_size; (void)d_ws; (void)ws_size;
    const float* params = (const float*)d_in[0];   // (N,7) float32
    const float* tmin   = (const float*)d_in[1];   // (N,)  float32
    float* out = (float*)d_out;                    // pos(N,3) ++ vel(N,3)
    int N = in_sizes[1];
    int blocks = (N + BLK - 1) / BLK;
    hipLaunchKernelGGL(sgp4_kernel, dim3(blocks), dim3(BLK), 0, stream,
                       params, tmin, out, N);
}